// LUSTER_Net_82892868813436
// MI455X (gfx1250) — compile-verified
//
#include <hip/hip_runtime.h>

#define L_  4
#define N_  50000
#define E_  800000
#define D_  128
#define H_  64
#define B_  16384
#define ED_ 256
#define K_  7

typedef float v2f __attribute__((ext_vector_type(2)));
typedef float v8f __attribute__((ext_vector_type(8)));

#define CDIV(a, b) (((a) + (b) - 1) / (b))

// ---------------- setup kernels ----------------

// conv_w [c][ci][k] -> wt2 [k][ci/2][c][2]  (K-pair interleave: one b64 per B-frag)
__global__ void k_transpose_w(const float* __restrict__ cw, float* __restrict__ wt2) {
    int idx = blockIdx.x * blockDim.x + threadIdx.x;
    if (idx >= ED_ * ED_ * K_) return;
    int c  = idx / (ED_ * K_);
    int r  = idx - c * (ED_ * K_);
    int ci = r / K_;
    int k  = r - ci * K_;
    wt2[(k * 128 + (ci >> 1)) * 512 + c * 2 + (ci & 1)] = cw[idx];
}

// W2 [l][k][n] -> W2P [l][k/2][n][2]
__global__ void k_w2p(const float* __restrict__ W2, float* __restrict__ w2p) {
    int idx = blockIdx.x * blockDim.x + threadIdx.x;
    if (idx >= L_ * H_ * D_) return;
    int l = idx / (H_ * D_);
    int r = idx - l * (H_ * D_);
    int k = r / D_;
    int n = r - k * D_;
    w2p[l * (H_ * D_) + (k >> 1) * 256 + n * 2 + (k & 1)] = W2[idx];
}

// seed layer_specific with per-sample b2 bias (both halves get b2[layer])
__global__ void k_ls_init(const int* __restrict__ nl, const float* __restrict__ b2,
                          float* __restrict__ LS) {
    int idx = blockIdx.x * blockDim.x + threadIdx.x;
    if (idx >= B_ * ED_) return;
    int b = idx >> 8;
    int c = idx & 255;
    LS[idx] = b2[nl[b] * D_ + (c & (D_ - 1))];
}

// ---------------- per-layer GCN kernels ----------------

__global__ void k_deg_init(float* __restrict__ deg, int* __restrict__ flags,
                           float* __restrict__ stats) {
    int v = blockIdx.x * blockDim.x + threadIdx.x;
    if (v < N_) { deg[v] = 1.0f; flags[v] = 0; }   // self-loop contributes 1
    if (v < 2) stats[v] = 0.0f;
}

__global__ void k_deg_accum(const int* __restrict__ ei_l, float* __restrict__ deg) {
    int e = blockIdx.x * blockDim.x + threadIdx.x;
    if (e >= E_) return;
    atomicAdd(&deg[ei_l[E_ + e]], 1.0f);
}

// deg -> dis (in place), and seed s with self-loop term dis^2 * x
__global__ void k_dis_sinit(const float* __restrict__ x_l, float* __restrict__ dis,
                            float* __restrict__ s) {
    int v = blockIdx.x * blockDim.x + threadIdx.x;
    if (v >= N_) return;
    float di = rsqrtf(fmaxf(dis[v], 1e-12f));
    dis[v] = di;
    s[v]   = di * di * x_l[v];
}

__global__ void k_s_accum(const int* __restrict__ ei_l, const float* __restrict__ dis,
                          const float* __restrict__ x_l, float* __restrict__ s) {
    int e = blockIdx.x * blockDim.x + threadIdx.x;
    if (e >= E_) return;
    int src = ei_l[e];
    int dst = ei_l[E_ + e];
    atomicAdd(&s[dst], dis[src] * dis[dst] * x_l[src]);
}

__global__ void k_stats(const float* __restrict__ s, float* __restrict__ stats) {
    float sum = 0.f, sq = 0.f;
    for (int v = blockIdx.x * blockDim.x + threadIdx.x; v < N_; v += gridDim.x * blockDim.x) {
        float t = s[v]; sum += t; sq += t * t;
    }
#pragma unroll
    for (int off = 16; off > 0; off >>= 1) {
        sum += __shfl_down(sum, off, 32);
        sq  += __shfl_down(sq,  off, 32);
    }
    __shared__ float red[16];
    int lane = threadIdx.x & 31, wid = threadIdx.x >> 5;
    if (lane == 0) { red[wid] = sum; red[8 + wid] = sq; }
    __syncthreads();
    if (threadIdx.x == 0) {
        float a = 0.f, b = 0.f;
        for (int i = 0; i < 8; i++) { a += red[i]; b += red[8 + i]; }
        atomicAdd(&stats[0], a);
        atomicAdd(&stats[1], b);
    }
}

// mark needed dst nodes for this layer
__global__ void k_flags(const int* __restrict__ nl, const int* __restrict__ ln,
                        const int* __restrict__ rn, int l, int* __restrict__ flags) {
    int b = blockIdx.x * blockDim.x + threadIdx.x;
    if (b >= B_) return;
    if (nl[b] == l) { flags[ln[b]] = 1; flags[rn[b]] = 1; }
}

// fold BatchNorm into 64 (a_j, c_j) coefficients; b1 cancels analytically
__global__ void k_coeff(const float* __restrict__ W1_l, const float* __restrict__ g_l,
                        const float* __restrict__ bt_l, const float* __restrict__ stats,
                        float* __restrict__ coef) {
    int j = threadIdx.x;
    if (j >= H_) return;
    float mu  = stats[0] * (1.0f / N_);
    float var = stats[1] * (1.0f / N_) - mu * mu;
    float w   = W1_l[j];
    coef[j]        = g_l[j] * w * rsqrtf(var * w * w + 1e-5f);  // a_j
    coef[H_ + j]   = bt_l[j];                                   // c_j
    if (j == 0) coef[2 * H_] = mu;
}

// agg[v,:] = self-loop term dis^2 * relu(a*(s-mu)+c), only at needed nodes
__global__ void k_agg_init(const int* __restrict__ flags, const float* __restrict__ s,
                           const float* __restrict__ dis, const float* __restrict__ coef,
                           float* __restrict__ agg) {
    int idx = blockIdx.x * blockDim.x + threadIdx.x;
    if (idx >= N_ * H_) return;
    int v = idx >> 6, j = idx & 63;
    if (!flags[v]) return;
    float t  = s[v] - coef[2 * H_];
    float h  = fmaxf(coef[j] * t + coef[H_ + j], 0.f);
    float di = dis[v];
    agg[idx] = di * di * h;
}

// one wave per edge; lane covers channels j=lane and j=lane+32; skip un-needed dst
__global__ void k_agg_edges(const int* __restrict__ ei_l, const int* __restrict__ flags,
                            const float* __restrict__ s, const float* __restrict__ dis,
                            const float* __restrict__ coef, float* __restrict__ agg) {
    int tid  = blockIdx.x * blockDim.x + threadIdx.x;
    int e    = tid >> 5;
    int lane = tid & 31;
    if (e >= E_) return;
    int dst = ei_l[E_ + e];
    if (!flags[dst]) return;
    int src    = ei_l[e];
    float t    = s[src] - coef[128];
    float norm = dis[src] * dis[dst];
    float h0 = fmaxf(coef[lane]      * t + coef[64 + lane], 0.f);
    float h1 = fmaxf(coef[32 + lane] * t + coef[96 + lane], 0.f);
    atomicAdd(&agg[(long)dst * 64 + lane],      norm * h0);
    atomicAdd(&agg[(long)dst * 64 + 32 + lane], norm * h1);
}

// LS[b, side*128 + 64*ntg ..] += (layer-masked agg[node]) @ W2[l]
// one wave = 16 samples x 64 cols (4 N-tiles), branch-free masked A
__global__ void k_ls(const int* __restrict__ nl, const int* __restrict__ ln,
                     const int* __restrict__ rn, const float* __restrict__ agg,
                     const float* __restrict__ w2p_l, int l, float* __restrict__ LS) {
    int tid  = blockIdx.x * blockDim.x + threadIdx.x;
    int w    = tid >> 5;
    int lane = tid & 31;
    int side = w & 1;
    int ntg  = (w >> 1) & 1;
    int tm   = w >> 2;
    if (tm >= B_ / 16) return;

    int mrow  = lane & 15;            // A: M index
    int khalf = (lane >> 4) << 1;     // A/B: K sub-offset 0 or 2
    int cn    = lane & 15;            // C/D, B: N index
    int cmb   = (lane >> 4) << 3;     // C/D: M base 0 or 8

    int b      = tm * 16 + mrow;
    float msk  = (nl[b] == l) ? 1.0f : 0.0f;
    int node   = side ? rn[b] : ln[b];           // always in [0,N)
    const float* arow = agg + (long)node * 64;

    float* lsb = LS + (long)(tm * 16) * ED_ + side * D_ + ntg * 64 + cn;
    v8f acc[4];
#pragma unroll
    for (int nt = 0; nt < 4; nt++)
#pragma unroll
        for (int i = 0; i < 8; i++) acc[nt][i] = lsb[nt * 16 + (cmb + i) * ED_];

#pragma unroll
    for (int k0 = 0; k0 < H_; k0 += 4) {
        v2f a;
        a[0] = arow[k0 + khalf]     * msk;
        a[1] = arow[k0 + khalf + 1] * msk;
        const float* bp = w2p_l + ((k0 + khalf) >> 1) * 256 + (ntg * 64 + cn) * 2;
#pragma unroll
        for (int nt = 0; nt < 4; nt++) {
            v2f bf;
            bf[0] = bp[nt * 32 + 0];
            bf[1] = bp[nt * 32 + 1];
            acc[nt] = __builtin_amdgcn_wmma_f32_16x16x4_f32(false, a, false, bf, (short)0,
                                                            acc[nt], false, false);
        }
    }
#pragma unroll
    for (int nt = 0; nt < 4; nt++)
#pragma unroll
        for (int i = 0; i < 8; i++) lsb[nt * 16 + (cmb + i) * ED_] = acc[nt][i];
}

// ---------------- conv1d over batch dim as 7 shifted GEMMs (f32 WMMA) ----------------
// one wave = 16 batch rows x 64 out-channels (4 N-tiles): 1 A-load feeds 4 WMMAs

__global__ void k_conv(const float* __restrict__ LS, const float* __restrict__ wt2,
                       const float* __restrict__ cb, float* __restrict__ sh) {
    int tid  = blockIdx.x * blockDim.x + threadIdx.x;
    int w    = tid >> 5;
    int lane = tid & 31;
    int ntg  = w & 3;                 // group of 4 N-tiles (64 out-channels)
    int tm   = w >> 2;
    if (tm >= B_ / 16) return;

    int mrow  = lane & 15;
    int khalf = (lane >> 4) << 1;
    int cn    = lane & 15;
    int cmb   = (lane >> 4) << 3;

    v8f acc[4];
#pragma unroll
    for (int nt = 0; nt < 4; nt++)
#pragma unroll
        for (int i = 0; i < 8; i++) acc[nt][i] = 0.f;

    for (int k = 0; k < K_; k++) {
        int g = tm * 16 + mrow + k - 3;               // zero-padded batch index
        float msk = (g >= 0 && g < B_) ? 1.0f : 0.0f;
        int gc = g < 0 ? 0 : (g >= B_ ? B_ - 1 : g);  // clamped: load always legal
        const float* arow = LS + (long)gc * ED_;
        const float* bkp  = wt2 + k * (128 * 512) + (ntg * 64 + cn) * 2;
        if (k + 1 < K_)   // prefetch next k-plane of weights (global_prefetch_b8)
            __builtin_prefetch(bkp + 128 * 512, 0, 0);
#pragma unroll 4
        for (int ci = 0; ci < ED_; ci += 4) {
            v2f a;
            a[0] = arow[ci + khalf]     * msk;
            a[1] = arow[ci + khalf + 1] * msk;
            const float* bp = bkp + ((ci + khalf) >> 1) * 512;
#pragma unroll
            for (int nt = 0; nt < 4; nt++) {
                v2f bf;
                bf[0] = bp[nt * 32 + 0];
                bf[1] = bp[nt * 32 + 1];
                acc[nt] = __builtin_amdgcn_wmma_f32_16x16x4_f32(false, a, false, bf,
                                                                (short)0, acc[nt],
                                                                false, false);
            }
        }
    }
#pragma unroll
    for (int nt = 0; nt < 4; nt++) {
        float bias = cb[ntg * 64 + nt * 16 + cn];
        float* ob  = sh + (long)(tm * 16) * ED_ + ntg * 64 + nt * 16 + cn;
#pragma unroll
        for (int i = 0; i < 8; i++) ob[(cmb + i) * ED_] = fmaxf(acc[nt][i] + bias, 0.f);
    }
}

// ---------------- ortho + heads: one wave per sample ----------------

__global__ void k_head(const float* __restrict__ LS, const float* __restrict__ sh,
                       const float* __restrict__ lcw, const float* __restrict__ lcb,
                       const float* __restrict__ lkw, const float* __restrict__ lkb,
                       float* __restrict__ out) {
    int tid  = blockIdx.x * blockDim.x + threadIdx.x;
    int b    = tid >> 5;
    int lane = tid & 31;
    if (b >= B_) return;
    const float* ls = LS + (long)b * ED_;
    const float* sv = sh + (long)b * ED_;
    float dot = 0, nrm = 0, d0 = 0, d1 = 0, d2 = 0, d3 = 0;
    float pl0 = 0, pl1 = 0, ps0 = 0, ps1 = 0;
    for (int c = lane; c < ED_; c += 32) {
        float lv = ls[c], s = sv[c];
        dot += lv * s;  nrm += s * s;
        d0 += s * lcw[c * 4 + 0]; d1 += s * lcw[c * 4 + 1];
        d2 += s * lcw[c * 4 + 2]; d3 += s * lcw[c * 4 + 3];
        pl0 += lv * lkw[c * 2 + 0]; pl1 += lv * lkw[c * 2 + 1];
        ps0 += s  * lkw[c * 2 + 0]; ps1 += s  * lkw[c * 2 + 1];
    }
#pragma unroll
    for (int off = 16; off > 0; off >>= 1) {
        dot += __shfl_down(dot, off, 32); nrm += __shfl_down(nrm, off, 32);
        d0  += __shfl_down(d0,  off, 32); d1  += __shfl_down(d1,  off, 32);
        d2  += __shfl_down(d2,  off, 32); d3  += __shfl_down(d3,  off, 32);
        pl0 += __shfl_down(pl0, off, 32); pl1 += __shfl_down(pl1, off, 32);
        ps0 += __shfl_down(ps0, off, 32); ps1 += __shfl_down(ps1, off, 32);
    }
    if (lane == 0) {
        float f = 1.0f - dot / (nrm + 1e-12f);     // enhanced = ls + (1-alpha)*shared
        out[b * 2 + 0] = pl0 + f * ps0 + lkb[0];
        out[b * 2 + 1] = pl1 + f * ps1 + lkb[1];
        float* dsc = out + (size_t)B_ * 2 + (size_t)b * 4;
        dsc[0] = d0 + lcb[0]; dsc[1] = d1 + lcb[1];
        dsc[2] = d2 + lcb[2]; dsc[3] = d3 + lcb[3];
    }
}

// ---------------- host ----------------

extern "C" void kernel_launch(void* const* d_in, const int* in_sizes, int n_in,
                              void* d_out, int out_size, void* d_ws, size_t ws_size,
                              hipStream_t stream) {
    (void)in_sizes; (void)n_in; (void)out_size; (void)ws_size;
    const float* x      = (const float*)d_in[0];
    const int*   ei     = (const int*)d_in[1];
    const float* W1     = (const float*)d_in[2];
    /* d_in[3] = b1: cancels inside BatchNorm, unused */
    const float* gamma  = (const float*)d_in[4];
    const float* beta   = (const float*)d_in[5];
    const float* W2     = (const float*)d_in[6];
    const float* b2     = (const float*)d_in[7];
    const float* conv_w = (const float*)d_in[8];
    const float* conv_b = (const float*)d_in[9];
    const float* lc_w   = (const float*)d_in[10];
    const float* lc_b   = (const float*)d_in[11];
    const float* link_w = (const float*)d_in[12];
    const float* link_b = (const float*)d_in[13];
    const int*   nl     = (const int*)d_in[14];
    const int*   ln     = (const int*)d_in[15];
    const int*   rn     = (const int*)d_in[16];
    float*       out    = (float*)d_out;

    // workspace carve-up (~49 MB)
    float* LS    = (float*)d_ws;                   // B*256
    float* SH    = LS + (size_t)B_ * ED_;          // B*256
    float* AGG   = SH + (size_t)B_ * ED_;          // N*64
    float* DIS   = AGG + (size_t)N_ * H_;          // N
    float* S     = DIS + N_;                       // N
    float* STATS = S + N_;                         // 2
    float* COEF  = STATS + 4;                      // 129 (+pad)
    float* WT2   = COEF + 136;                     // 7*256*256 (K-pair packed)
    float* W2P   = WT2 + (size_t)K_ * ED_ * ED_;   // L*H*D   (K-pair packed)
    int*   FLAGS = (int*)(W2P + (size_t)L_ * H_ * D_);   // N ints

    k_transpose_w<<<CDIV(ED_ * ED_ * K_, 256), 256, 0, stream>>>(conv_w, WT2);
    k_w2p<<<CDIV(L_ * H_ * D_, 256), 256, 0, stream>>>(W2, W2P);
    k_ls_init<<<CDIV(B_ * ED_, 256), 256, 0, stream>>>(nl, b2, LS);

    for (int l = 0; l < L_; l++) {
        const int*   ei_l = ei + (size_t)l * 2 * E_;
        const float* x_l  = x + (size_t)l * N_;
        k_deg_init<<<CDIV(N_, 256), 256, 0, stream>>>(DIS, FLAGS, STATS);
        k_deg_accum<<<CDIV(E_, 256), 256, 0, stream>>>(ei_l, DIS);
        k_dis_sinit<<<CDIV(N_, 256), 256, 0, stream>>>(x_l, DIS, S);
        k_s_accum<<<CDIV(E_, 256), 256, 0, stream>>>(ei_l, DIS, x_l, S);
        k_stats<<<CDIV(N_, 256), 256, 0, stream>>>(S, STATS);
        k_flags<<<CDIV(B_, 256), 256, 0, stream>>>(nl, ln, rn, l, FLAGS);
        k_coeff<<<1, 64, 0, stream>>>(W1 + l * H_, gamma + l * H_, beta + l * H_,
                                      STATS, COEF);
        k_agg_init<<<CDIV(N_ * H_, 256), 256, 0, stream>>>(FLAGS, S, DIS, COEF, AGG);
        k_agg_edges<<<CDIV(E_ * 32, 256), 256, 0, stream>>>(ei_l, FLAGS, S, DIS, COEF, AGG);
        k_ls<<<(B_ / 16) * 4 * 32 / 256, 256, 0, stream>>>(nl, ln, rn, AGG,
                                                           W2P + (size_t)l * H_ * D_, l, LS);
    }

    k_conv<<<(B_ / 16) * 4 * 32 / 256, 256, 0, stream>>>(LS, WT2, conv_b, SH);
    k_head<<<CDIV(B_ * 32, 256), 256, 0, stream>>>(LS, SH, lc_w, lc_b, link_w, link_b, out);
}